// SimpleConstellationMapper_29351806501268
// MI455X (gfx1250) — compile-verified
//
#include <hip/hip_runtime.h>

// ---------------------------------------------------------------------------
// SimpleConstellationMapper on MI455X (gfx1250)
//
// Roofline: 201 MB read + 67 MB write, ~0 FLOPs -> pure HBM-bound (~11.5 us at
// 23.3 TB/s). WMMA is inapplicable (6-bit pack + 64-entry gather, not a GEMM).
// Optimization = data movement:
//   * GLOBAL_LOAD_ASYNC_TO_LDS_B128: each wave stages 1536 B of bit data into
//     LDS with 512 B-contiguous-per-instruction coalescing (ASYNCcnt path).
//   * 64-entry normalized complex LUT built in LDS by wave 0 using a wave32
//     __shfl_xor butterfly reduction + v_rsq_f32.
//   * Per thread: 2 symbols, ds_load_b128 x3, LUT gather, global_store_b128.
// ---------------------------------------------------------------------------

#define M_BITS        6
#define SYMS_PER_THR  2
#define TPB           256                                  // 8 waves (wave32)
#define BYTES_PER_SYM (M_BITS * 4)                         // 24
#define WAVE_BYTES    (32 * SYMS_PER_THR * BYTES_PER_SYM)  // 1536
#define BLOCK_BYTES   ((TPB / 32) * WAVE_BYTES)            // 12288

typedef __attribute__((address_space(3))) unsigned char lds_byte_t;

__global__ __launch_bounds__(TPB, 1) void constellation_map_kernel(
    const int*   __restrict__ bits,
    const float* __restrict__ weights,
    float*       __restrict__ out)
{
    __shared__ float2 lut[64];
    __shared__ __align__(16) unsigned char stage[BLOCK_BYTES];

    const int tid  = threadIdx.x;
    const int lane = tid & 31;
    const int wv   = tid >> 5;

    // -- 1. async-stage this wave's 1536 B of bit data into LDS (coalesced) --
    {
        const unsigned char* gp = (const unsigned char*)bits
            + (size_t)blockIdx.x * BLOCK_BYTES
            + (size_t)wv * WAVE_BYTES
            + (size_t)lane * 16;
        unsigned lds_off = (unsigned)(unsigned long long)
            (lds_byte_t*)(stage + wv * WAVE_BYTES + lane * 16);
        // INST_OFFSET applies to both the LDS and the global address
        // (CDNA5 ISA), so one addr pair serves 3 ops.
        asm volatile(
            "global_load_async_to_lds_b128 %0, %1, off\n\t"
            "global_load_async_to_lds_b128 %0, %1, off offset:512\n\t"
            "global_load_async_to_lds_b128 %0, %1, off offset:1024"
            :: "v"(lds_off), "v"(gp) : "memory");
    }

    // -- 2. wave 0 builds the energy-normalized 64-entry complex LUT --------
    if (wv == 0) {
        const float2* w2 = (const float2*)weights;
        float2 w0 = w2[2 * lane + 0];
        float2 w1 = w2[2 * lane + 1];
        float e = w0.x * w0.x + w0.y * w0.y + w1.x * w1.x + w1.y * w1.y;
        #pragma unroll
        for (int m = 16; m >= 1; m >>= 1)       // wave32 butterfly reduction
            e += __shfl_xor(e, m, 32);
        float s = rsqrtf(e * (1.0f / 64.0f));   // rsqrt(mean energy)
        lut[2 * lane + 0] = make_float2(w0.x * s, w0.y * s);
        lut[2 * lane + 1] = make_float2(w1.x * s, w1.y * s);
    }
    __syncthreads();                            // LUT visible to all waves
    asm volatile("s_wait_asynccnt 0x0" ::: "memory");  // own stage data landed

    // -- 3. read 2 symbols (48 B) from LDS, pack MSB-first indices, gather --
    const int4* sp = (const int4*)(stage + (size_t)tid * (SYMS_PER_THR * BYTES_PER_SYM));
    int4 p0 = sp[0];
    int4 p1 = sp[1];
    int4 p2 = sp[2];

    int idx0 = (p0.x << 5) | (p0.y << 4) | (p0.z << 3)
             | (p0.w << 2) | (p1.x << 1) |  p1.y;
    int idx1 = (p1.z << 5) | (p1.w << 4) | (p2.x << 3)
             | (p2.y << 2) | (p2.z << 1) |  p2.w;

    float2 c0 = lut[idx0];
    float2 c1 = lut[idx1];

    // 2 complex64 -> one fully coalesced global_store_b128 per thread
    ((float4*)out)[(size_t)blockIdx.x * TPB + tid] =
        make_float4(c0.x, c0.y, c1.x, c1.y);
}

extern "C" void kernel_launch(void* const* d_in, const int* in_sizes, int n_in,
                              void* d_out, int out_size, void* d_ws, size_t ws_size,
                              hipStream_t stream)
{
    (void)n_in; (void)out_size; (void)d_ws; (void)ws_size;

    const int*   bits    = (const int*)d_in[0];    // [BATCH, 6] int32 in {0,1}
    const float* weights = (const float*)d_in[1];  // [64, 2] float32
    float*       out     = (float*)d_out;          // [BATCH] complex64 (re,im)

    const int batch   = in_sizes[0] / M_BITS;      // 8388608
    const int threads = batch / SYMS_PER_THR;      // 4194304
    const int blocks  = threads / TPB;             // 16384

    constellation_map_kernel<<<blocks, TPB, 0, stream>>>(bits, weights, out);
}